// Convpass_swin_hypernet_fuse_pos_22359599743392
// MI455X (gfx1250) — compile-verified
//
#include <hip/hip_runtime.h>
#include <hip/hip_bf16.h>

typedef __attribute__((ext_vector_type(2))) float v2f;
typedef __attribute__((ext_vector_type(8))) float v8f;

#define WMMA_F32(a, b, c) \
    __builtin_amdgcn_wmma_f32_16x16x4_f32(false, (a), false, (b), (short)0, (c), false, false)

__device__ __forceinline__ float qgelu(float x) {
    // x * sigmoid(1.702 x)
    return x / (1.0f + __expf(-1.702f * x));
}

// A fragment: row-major src; element (row0+l, k0+2*hi+v)
__device__ __forceinline__ v2f load_afrag(const float* __restrict__ A, int lda,
                                          int row0, int k0, int l, int hi) {
    const float* p = A + (size_t)(row0 + l) * lda + (k0 + 2 * hi);
    v2f a; a.x = p[0]; a.y = p[1];
    return a;
}
// B fragment: row-major K x N src; element (k0+2*hi+v, n0+l)
__device__ __forceinline__ v2f load_bfrag(const float* __restrict__ B, int ldb,
                                          int k0, int n0, int l, int hi) {
    const float* p = B + (size_t)(k0 + 2 * hi) * ldb + (n0 + l);
    v2f b; b.x = p[0]; b.y = p[ldb];
    return b;
}

// ---------------------------------------------------------------------------
// Stage 1: hbar[b][64] = mean over 784 positions of relu(x @ Wm1 + bm1)
// grid (49, 64), block 128 (4 waves, one 16-col N-tile each)
// ---------------------------------------------------------------------------
__global__ void k_meta_down(const float* __restrict__ x, const float* __restrict__ Wm1,
                            const float* __restrict__ bm1, float* __restrict__ hbar) {
    const int lane = threadIdx.x & 31;
    const int wave = threadIdx.x >> 5;      // 0..3 -> N-tile
    const int l = lane & 15, hi = lane >> 4;
    const int b  = blockIdx.y;
    const int mt = blockIdx.x;              // 0..48 (784 = 49*16, tile within one b)
    const int row0 = b * 784 + mt * 16;
    const int n0 = wave * 16;

    v8f c = {};
    for (int k = 0; k < 384; k += 4) {
        v2f a  = load_afrag(x, 384, row0, k, l, hi);
        v2f bb = load_bfrag(Wm1, 64, k, n0, l, hi);
        c = WMMA_F32(a, bb, c);
    }
    const float bias = bm1[n0 + l];
    float s = 0.f;
#pragma unroll
    for (int r = 0; r < 8; ++r) {
        float v = c[r] + bias;
        s += (v > 0.f) ? v : 0.f;
    }
    s += __shfl_xor(s, 16, 32);             // combine the two wave32 halves (rows 0-7 / 8-15)
    if (hi == 0)
        atomicAdd(hbar + b * 64 + n0 + l, s * (1.0f / 784.0f));
}

// ---------------------------------------------------------------------------
// Stage 2: p[b][64] = emb + bm2 + hbar[b] @ Wm2     (tiny: VALU)
// grid 16, block 256
// ---------------------------------------------------------------------------
__global__ void k_hyper_p(const float* __restrict__ hbar, const float* __restrict__ Wm2,
                          const float* __restrict__ bm2, const float* __restrict__ emb,
                          float* __restrict__ p) {
    int t = blockIdx.x * 256 + threadIdx.x;
    int b = t >> 6, e = t & 63;
    float acc = bm2[e] + emb[e];
    const float* hb = hbar + b * 64;
#pragma unroll 8
    for (int k = 0; k < 64; ++k) acc += hb[k] * Wm2[k * 64 + e];
    p[t] = acc;
}

// ---------------------------------------------------------------------------
// Stage 3: wflat = p @ Wh + bh, scatter-stored into wT[b][tap][din][dout]
//          (n = (dout*96 + din)*9 + tap)
// grid (648, 4), block 256 (8 waves, one N-tile each; 648*8 = 5184 = 82944/16)
// ---------------------------------------------------------------------------
__global__ void k_hyper_w(const float* __restrict__ p, const float* __restrict__ Wh,
                          const float* __restrict__ bh, float* __restrict__ wT) {
    const int lane = threadIdx.x & 31;
    const int wave = threadIdx.x >> 5;
    const int l = lane & 15, hi = lane >> 4;
    const int nt = blockIdx.x * 8 + wave;   // 0..5183
    const int mt = blockIdx.y;              // 0..3 (64 batches / 16)
    const int n0 = nt * 16;

    v8f c = {};
    for (int k = 0; k < 64; k += 4) {
        v2f a  = load_afrag(p, 64, mt * 16, k, l, hi);
        v2f bb = load_bfrag(Wh, 82944, k, n0, l, hi);
        c = WMMA_F32(a, bb, c);
    }
    const int n = n0 + l;
    const float bias = bh[n];
    const int dout = n / 864;
    const int rem  = n - dout * 864;
    const int din  = rem / 9;
    const int tap  = rem - din * 9;
#pragma unroll
    for (int r = 0; r < 8; ++r) {
        int bb_ = mt * 16 + r + 8 * hi;
        wT[(((size_t)bb_ * 9 + tap) * 96 + din) * 96 + dout] = c[r] + bias;
    }
}

// ---------------------------------------------------------------------------
// Stage 4: xdp[b][1+h][1+w][d] = quickgelu(x @ Wd + bd)  (zero-padded 30x30)
// grid 3136 (= 50176/16), block 192 (6 waves, one 16-col N-tile each; N=96)
// ---------------------------------------------------------------------------
__global__ void k_down(const float* __restrict__ x, const float* __restrict__ Wd,
                       const float* __restrict__ bd, float* __restrict__ xdp) {
    const int lane = threadIdx.x & 31;
    const int wave = threadIdx.x >> 5;      // 0..5
    const int l = lane & 15, hi = lane >> 4;
    const int mt = blockIdx.x;
    const int row0 = mt * 16;
    const int n0 = wave * 16;

    v8f c = {};
    for (int k = 0; k < 384; k += 4) {
        v2f a  = load_afrag(x, 384, row0, k, l, hi);
        v2f bb = load_bfrag(Wd, 96, k, n0, l, hi);
        c = WMMA_F32(a, bb, c);
    }
    const float bias = bd[n0 + l];
#pragma unroll
    for (int r = 0; r < 8; ++r) {
        int g = row0 + r + 8 * hi;
        int b = g / 784;
        int pos = g - b * 784;
        int h = pos / 28;
        int w = pos - h * 28;
        size_t dst = (((size_t)b * 30 + (h + 1)) * 30 + (w + 1)) * 96 + n0 + l;
        xdp[dst] = qgelu(c[r] + bias);
    }
}

// ---------------------------------------------------------------------------
// Stage 5: per-sample dynamic 3x3 conv as 9 tap-GEMMs + quickgelu
// y[b][pos][dout] = qgelu( sum_{tap,din} xdp[b][h+dh][w+dw][din] * wT[b][tap][din][dout] )
// grid (49, 64), block 192 (6 waves, one dout N-tile each)
// ---------------------------------------------------------------------------
__global__ void k_dynconv(const float* __restrict__ xdp, const float* __restrict__ wT,
                          float* __restrict__ y) {
    const int lane = threadIdx.x & 31;
    const int wave = threadIdx.x >> 5;
    const int l = lane & 15, hi = lane >> 4;
    const int b  = blockIdx.y;
    const int mt = blockIdx.x;
    const int n0 = wave * 16;
    const int pos = mt * 16 + l;            // this lane's A-row position
    const int h = pos / 28;
    const int w = pos - h * 28;
    const float* xb = xdp + (size_t)b * 900 * 96;
    const float* wb = wT  + (size_t)b * 9 * 96 * 96;

    v8f c = {};
    for (int tap = 0; tap < 9; ++tap) {
        int dh = tap / 3, dw = tap - dh * 3;
        // padded coords: (h + dh, w + dw) covers (h-1..h+1)+1 etc.
        const float* arow = xb + (((h + dh) * 30) + (w + dw)) * 96 + 2 * hi;
        const float* bmat = wb + tap * 96 * 96 + (2 * hi) * 96 + n0 + l;
#pragma unroll 4
        for (int k = 0; k < 96; k += 4) {
            v2f a;  a.x  = arow[k];          a.y  = arow[k + 1];
            v2f bb; bb.x = bmat[k * 96];     bb.y = bmat[k * 96 + 96];
            c = WMMA_F32(a, bb, c);
        }
    }
#pragma unroll
    for (int r = 0; r < 8; ++r) {
        int pr = mt * 16 + r + 8 * hi;
        y[((size_t)b * 784 + pr) * 96 + n0 + l] = qgelu(c[r]);
    }
}

// ---------------------------------------------------------------------------
// Stage 6: out = y @ Wu + bu
// grid (3136, 3), block 256 (8 waves; n_tile = blockIdx.y*8 + wave, 24 total)
// ---------------------------------------------------------------------------
__global__ void k_up(const float* __restrict__ y, const float* __restrict__ Wu,
                     const float* __restrict__ bu, float* __restrict__ out) {
    const int lane = threadIdx.x & 31;
    const int wave = threadIdx.x >> 5;
    const int l = lane & 15, hi = lane >> 4;
    const int mt = blockIdx.x;
    const int nt = blockIdx.y * 8 + wave;   // 0..23
    const int row0 = mt * 16;
    const int n0 = nt * 16;

    v8f c = {};
    for (int k = 0; k < 96; k += 4) {
        v2f a  = load_afrag(y, 96, row0, k, l, hi);
        v2f bb = load_bfrag(Wu, 384, k, n0, l, hi);
        c = WMMA_F32(a, bb, c);
    }
    const float bias = bu[n0 + l];
#pragma unroll
    for (int r = 0; r < 8; ++r) {
        int g = row0 + r + 8 * hi;
        out[(size_t)g * 384 + n0 + l] = c[r] + bias;
    }
}

// ---------------------------------------------------------------------------
extern "C" void kernel_launch(void* const* d_in, const int* in_sizes, int n_in,
                              void* d_out, int out_size, void* d_ws, size_t ws_size,
                              hipStream_t stream) {
    const float* x   = (const float*)d_in[0];
    const float* Wd  = (const float*)d_in[1];
    const float* bd  = (const float*)d_in[2];
    const float* Wm1 = (const float*)d_in[3];
    const float* bm1 = (const float*)d_in[4];
    const float* Wm2 = (const float*)d_in[5];
    const float* bm2 = (const float*)d_in[6];
    const float* Wh  = (const float*)d_in[7];
    const float* bh  = (const float*)d_in[8];
    const float* emb = (const float*)d_in[9];
    const float* Wu  = (const float*)d_in[10];
    const float* bu  = (const float*)d_in[11];
    float* out = (float*)d_out;

    // workspace layout
    const size_t off_hbar = 0;                       // 64*64*4          = 16 KB
    const size_t off_xdp  = off_hbar + 16384;        // 64*30*30*96*4    = 22,118,400
    const size_t off_wT   = off_xdp + 22118400;      // 64*9*96*96*4     = 21,233,664
    const size_t off_p    = off_wT  + 21233664;      // 64*64*4          = 16 KB
    const size_t off_y    = off_p   + 16384;         // 64*784*96*4      = 19,267,584
    char* ws = (char*)d_ws;
    float* hbar = (float*)(ws + off_hbar);
    float* xdp  = (float*)(ws + off_xdp);
    float* wT   = (float*)(ws + off_wT);
    float* p    = (float*)(ws + off_p);
    float* yb   = (float*)(ws + off_y);

    // zero hbar (atomic accumulation) + xdp (padding borders); graph-capture safe
    hipMemsetAsync(d_ws, 0, off_xdp + 22118400, stream);

    k_meta_down<<<dim3(49, 64), 128, 0, stream>>>(x, Wm1, bm1, hbar);
    k_hyper_p  <<<16, 256, 0, stream>>>(hbar, Wm2, bm2, emb, p);
    k_hyper_w  <<<dim3(648, 4), 256, 0, stream>>>(p, Wh, bh, wT);
    k_down     <<<3136, 192, 0, stream>>>(x, Wd, bd, xdp);
    k_dynconv  <<<dim3(49, 64), 192, 0, stream>>>(xdp, wT, yb);
    k_up       <<<dim3(3136, 3), 256, 0, stream>>>(yb, Wu, bu, out);
}